// AdverCETime_39994735460870
// MI455X (gfx1250) — compile-verified
//
#include <hip/hip_runtime.h>
#include <math.h>

// ---------------------------------------------------------------------------
// AdverCETime loss for MI455X (gfx1250, wave32).
//
//  key[b,v]  = noise_logits[b,v] + gumbel[b,v]   (same ranking as logp+gumbel)
//  neg set   = top-K(key, v != target)           (set only; LSE order-free)
//  type_loss = mean_b( log(sum exp(p_scores[sel+tgt])) - p_scores[tgt] )
//  time_loss = mean_b( ((pred - (tt - last))/4320)^2 ) / 5
//
// One workgroup (512 thr = 16 wave32) per row.
//  pass A: async-staged (GLOBAL_LOAD_ASYNC_TO_LDS_B128, double-buffered LDS,
//          ASYNCcnt pipeline) 12-bit LDS histogram of ordered keys
//  pass B: direct streaming (L2-resident) accumulate exp(p) above threshold
//          bin; exact O(n^2) rank select inside the tie bin
// ---------------------------------------------------------------------------

#define KSEL        1024
#define GRAN        4320.0f
#define TPB         512
#define NBINS       4096             // top 12 bits of ordered key
#define BIN_SHIFT   20               // 32 - 12
#define BINS_PER_CHUNK (NBINS / TPB) // 8
#define CAND_CAP    3072             // tie-bin candidates (expected ~512)
#define POOL_U32    8192             // 32 KB overlay: pass A staging / pass B cand

#if __has_builtin(__builtin_amdgcn_global_load_async_to_lds_b128) && \
    __has_builtin(__builtin_amdgcn_s_wait_asynccnt)
#define HAVE_ASYNC_LDS 1
#endif

typedef int v4i __attribute__((ext_vector_type(4)));
typedef __attribute__((address_space(1))) v4i as1_v4i;   // global int4
typedef __attribute__((address_space(3))) v4i as3_v4i;   // LDS int4

__device__ __forceinline__ unsigned int ord32(float f) {
    // order-preserving map: f0 < f1  <=>  ord32(f0) < ord32(f1)  (as uint)
    unsigned int i = __float_as_uint(f);
    return i ^ ((unsigned int)(((int)i) >> 31) | 0x80000000u);
}

__global__ __launch_bounds__(TPB) void adver_row_kernel(
    const float* __restrict__ noise,      // [B,V]
    const float* __restrict__ gum,        // [B,V]
    const float* __restrict__ psc,        // [B,V]
    const float* __restrict__ pred,       // [B,1]
    const float* __restrict__ tseq,       // [B,L]
    const float* __restrict__ ttime,      // [B]
    const int*   __restrict__ tgt_id,     // [B]
    const int*   __restrict__ seqlen,     // [B]
    float*       __restrict__ row_out,    // [B] workspace
    int V, int L)
{
    __shared__ unsigned int hist[NBINS];      // 16 KB
    __shared__ unsigned int pool[POOL_U32];   // 32 KB (staging <-> candidates)
    __shared__ unsigned int chunk_sum[TPB];   //  2 KB
    __shared__ float        redf[TPB];        //  2 KB
    __shared__ unsigned int s_tb, s_need, s_cand_cnt;

    const int b   = blockIdx.x;
    const int tid = threadIdx.x;
    const int tgt = tgt_id[b];

    const float* __restrict__ nrow = noise + (size_t)b * V;
    const float* __restrict__ grow = gum   + (size_t)b * V;
    const float* __restrict__ prow = psc   + (size_t)b * V;
    const float4* __restrict__ nv4 = (const float4*)nrow;
    const float4* __restrict__ gv4 = (const float4*)grow;
    const int V4 = V >> 2;

    // ---- zero histogram ----------------------------------------------------
    for (int i = tid; i < NBINS; i += TPB) hist[i] = 0u;
    if (tid == 0) s_cand_cnt = 0u;
    __syncthreads();

    // ---- pass A: histogram of ordered keys (skip target column) -----------
#ifdef HAVE_ASYNC_LDS
    {
        // Double-buffered async staging: each of the 512 lanes copies one
        // 16-byte chunk of noise and one of gumbel per tile (8 KB + 8 KB per
        // buffer).  Every wave always issues exactly 2 async loads per stage
        // (tail lanes clamp their address) so s_wait_asynccnt accounting is
        // uniform across waves.
        float4* poolF4 = (float4*)pool;
        const int nTiles = (V4 + TPB - 1) / TPB;

        auto stage = [&](int t, int buf) {
            int chunk = t * TPB + tid;
            if (chunk >= V4) chunk = V4 - 1;           // clamp, keep wave active
            const float4* gn = nv4 + chunk;
            const float4* gg = gv4 + chunk;
            float4* ln = poolF4 + buf * 1024 + tid;          // noise slot
            float4* lg = poolF4 + buf * 1024 + 512 + tid;    // gumbel slot
            __builtin_amdgcn_global_load_async_to_lds_b128(
                (as1_v4i*)(void*)gn, (as3_v4i*)(void*)ln, 0, 0);
            __builtin_amdgcn_global_load_async_to_lds_b128(
                (as1_v4i*)(void*)gg, (as3_v4i*)(void*)lg, 0, 0);
        };

        if (nTiles > 0) stage(0, 0);
        for (int t = 0; t < nTiles; ++t) {
            const int buf = t & 1;
            if (t + 1 < nTiles) {
                stage(t + 1, (t + 1) & 1);
                __builtin_amdgcn_s_wait_asynccnt(2);   // tile t landed
            } else {
                __builtin_amdgcn_s_wait_asynccnt(0);
            }
            __syncthreads();                           // all waves' data ready
            int chunk = t * TPB + tid;
            if (chunk < V4) {
                float4 n4 = poolF4[buf * 1024 + tid];
                float4 g4 = poolF4[buf * 1024 + 512 + tid];
                int v0 = chunk << 2;
                float k0 = n4.x + g4.x, k1 = n4.y + g4.y;
                float k2 = n4.z + g4.z, k3 = n4.w + g4.w;
                if (v0 + 0 != tgt) atomicAdd(&hist[ord32(k0) >> BIN_SHIFT], 1u);
                if (v0 + 1 != tgt) atomicAdd(&hist[ord32(k1) >> BIN_SHIFT], 1u);
                if (v0 + 2 != tgt) atomicAdd(&hist[ord32(k2) >> BIN_SHIFT], 1u);
                if (v0 + 3 != tgt) atomicAdd(&hist[ord32(k3) >> BIN_SHIFT], 1u);
            }
            __syncthreads();                           // before buf reuse
        }
    }
#else
    for (int i = tid; i < V4; i += TPB) {
        float4 n4 = nv4[i];
        float4 g4 = gv4[i];
        int v0 = i << 2;
        float k0 = n4.x + g4.x, k1 = n4.y + g4.y;
        float k2 = n4.z + g4.z, k3 = n4.w + g4.w;
        if (v0 + 0 != tgt) atomicAdd(&hist[ord32(k0) >> BIN_SHIFT], 1u);
        if (v0 + 1 != tgt) atomicAdd(&hist[ord32(k1) >> BIN_SHIFT], 1u);
        if (v0 + 2 != tgt) atomicAdd(&hist[ord32(k2) >> BIN_SHIFT], 1u);
        if (v0 + 3 != tgt) atomicAdd(&hist[ord32(k3) >> BIN_SHIFT], 1u);
    }
#endif
    for (int v = (V4 << 2) + tid; v < V; v += TPB) {
        if (v != tgt) atomicAdd(&hist[ord32(nrow[v] + grow[v]) >> BIN_SHIFT], 1u);
    }
    __syncthreads();

    // ---- find threshold bin TB: count(bin > TB) < K <= count(bin >= TB) ---
    {
        unsigned int cs = 0u;
        int base = tid * BINS_PER_CHUNK;
        #pragma unroll
        for (int k = 0; k < BINS_PER_CHUNK; ++k) cs += hist[base + k];
        chunk_sum[tid] = cs;
    }
    __syncthreads();
    if (tid == 0) {
        unsigned int run = 0u;
        int tb = 0; unsigned int above = 0u;
        for (int c = TPB - 1; c >= 0; --c) {
            unsigned int cs = chunk_sum[c];
            if (run + cs >= (unsigned int)KSEL) {
                unsigned int a = run;
                for (int bb = (c + 1) * BINS_PER_CHUNK - 1; bb >= c * BINS_PER_CHUNK; --bb) {
                    unsigned int h = hist[bb];
                    if (a + h >= (unsigned int)KSEL) { tb = bb; above = a; break; }
                    a += h;
                }
                break;
            }
            run += cs;
        }
        s_tb   = (unsigned int)tb;
        s_need = (unsigned int)KSEL - above;
    }
    __syncthreads();

    const unsigned int TB = s_tb;
    unsigned int* __restrict__ cu   = pool;              // overlay staging pool
    unsigned int* __restrict__ cidx = pool + CAND_CAP;

    // ---- pass B: accumulate exp(p) above TB; collect tie-bin candidates ---
    float psum = 0.0f;
    for (int i = tid; i < V4; i += TPB) {
        __builtin_prefetch(&nv4[i + TPB], 0, 1);   // global_prefetch_b8
        __builtin_prefetch(&gv4[i + TPB], 0, 1);
        float4 n4 = nv4[i];
        float4 g4 = gv4[i];
        int v0 = i << 2;
        float kk[4] = { n4.x + g4.x, n4.y + g4.y, n4.z + g4.z, n4.w + g4.w };
        #pragma unroll
        for (int j = 0; j < 4; ++j) {
            int v = v0 + j;
            if (v == tgt) continue;
            unsigned int u   = ord32(kk[j]);
            unsigned int bin = u >> BIN_SHIFT;
            if (bin > TB) {
                psum += expf(prow[v]);
            } else if (bin == TB) {
                unsigned int c = atomicAdd(&s_cand_cnt, 1u);
                if (c < CAND_CAP) { cu[c] = u; cidx[c] = (unsigned int)v; }
            }
        }
    }
    for (int v = (V4 << 2) + tid; v < V; v += TPB) {
        if (v == tgt) continue;
        unsigned int u   = ord32(nrow[v] + grow[v]);
        unsigned int bin = u >> BIN_SHIFT;
        if (bin > TB) {
            psum += expf(prow[v]);
        } else if (bin == TB) {
            unsigned int c = atomicAdd(&s_cand_cnt, 1u);
            if (c < CAND_CAP) { cu[c] = u; cidx[c] = (unsigned int)v; }
        }
    }
    __syncthreads();

    // ---- exact select of the remaining `need` entries inside the tie bin --
    {
        unsigned int n    = s_cand_cnt; if (n > CAND_CAP) n = CAND_CAP;
        unsigned int need = s_need;
        for (unsigned int j = tid; j < n; j += TPB) {
            unsigned int uj = cu[j], ij = cidx[j];
            unsigned int rank = 0u;
            for (unsigned int i2 = 0; i2 < n; ++i2) {
                unsigned int ui = cu[i2];
                rank += (ui > uj) || (ui == uj && cidx[i2] < ij);
            }
            if (rank < need) psum += expf(prow[ij]);
        }
    }
    const float s0 = prow[tgt];
    if (tid == 0) psum += expf(s0);   // target score participates in LSE

    // ---- block reduction (deterministic tree) -----------------------------
    redf[tid] = psum;
    __syncthreads();
    for (int s = TPB / 2; s > 0; s >>= 1) {
        if (tid < s) redf[tid] += redf[tid + s];
        __syncthreads();
    }

    if (tid == 0) {
        float lse = logf(redf[0]);
        int len = seqlen[b];
        if (len < 1) len = 1;
        if (len > L) len = L;
        float last_t  = tseq[(size_t)b * L + (len - 1)];
        float tgt_int = ttime[b] - last_t;
        float d = (pred[b] - tgt_int) / GRAN;
        row_out[b] = (lse - s0) + d * d * 0.2f;   // per-row contribution
    }
}

__global__ __launch_bounds__(TPB) void adver_reduce_kernel(
    const float* __restrict__ in, float* __restrict__ out, int n)
{
    __shared__ float r[TPB];
    float s = 0.0f;
    for (int i = threadIdx.x; i < n; i += TPB) s += in[i];
    r[threadIdx.x] = s;
    __syncthreads();
    for (int st = TPB / 2; st > 0; st >>= 1) {
        if (threadIdx.x < st) r[threadIdx.x] += r[threadIdx.x + st];
        __syncthreads();
    }
    if (threadIdx.x == 0) out[0] = r[0] / (float)n;
}

extern "C" void kernel_launch(void* const* d_in, const int* in_sizes, int n_in,
                              void* d_out, int out_size, void* d_ws, size_t ws_size,
                              hipStream_t stream) {
    const float* noise = (const float*)d_in[0];
    const float* psc   = (const float*)d_in[1];
    const float* pred  = (const float*)d_in[2];
    const float* tseq  = (const float*)d_in[3];
    const float* tt    = (const float*)d_in[4];
    const float* gum   = (const float*)d_in[5];
    const int*   tgt   = (const int*)d_in[6];
    const int*   slen  = (const int*)d_in[7];

    const int B = in_sizes[6];            // target_id count
    const int V = in_sizes[0] / B;
    const int L = in_sizes[3] / B;

    float* row_out = (float*)d_ws;        // B floats of scratch

    adver_row_kernel<<<B, TPB, 0, stream>>>(noise, gum, psc, pred, tseq, tt,
                                            tgt, slen, row_out, V, L);
    adver_reduce_kernel<<<1, TPB, 0, stream>>>(row_out, (float*)d_out, B);
}